// BlockSelfAttentionModule_4750233830020
// MI455X (gfx1250) — compile-verified
//
#include <hip/hip_runtime.h>

// ---------------------------------------------------------------------------
// out[g,l,m] = sum_e q[g,l,e] * ( r_voice[l%8, m%8, e, h] + T[l/8, m/8, e, h] )
//   g = b*8+h, T[fi,fj] = e_past[fi-fj] (fj<=fi) else e_future[fj-fi]
// Since the voice term only depends on m%8 and the time term only on m/8,
// compute P1 (32x64) and P2 (32x48) per (h,fi) block with fp32 WMMA, then
// broadcast-add into the 18.9MB output (the true bottleneck at 23.3 TB/s).
// ---------------------------------------------------------------------------

typedef __attribute__((ext_vector_type(2))) float v2f;
typedef __attribute__((ext_vector_type(4))) float v4f;
typedef __attribute__((ext_vector_type(8))) float v8f;

#define EDIM 16
#define NH   8
#define NV   8     // DI == DO == 8 voices
#define NF   48
#define LDIM 384   // NF * NV
#define STR  116   // padded LDS row stride (112 data cols + pad)

__global__ __launch_bounds__(256) void relattn_fused(
    const float* __restrict__ q,        // (32, 384, 16)
    const float* __restrict__ r_voice,  // (8, 8, 16, 8)   [vi, vo, e, h]
    const float* __restrict__ e_past,   // (48, 16, 8)     [d, e, h]
    const float* __restrict__ e_future, // (48, 16, 8)
    float* __restrict__ out)            // (32, 384, 384)
{
    __shared__ float lds[32 * STR];

    const int blk  = blockIdx.x;   // 0..383
    const int h    = blk & 7;      // head
    const int fi   = blk >> 3;     // input frame 0..47

    const int tid  = threadIdx.x;
    const int wave = tid >> 5;     // wave-uniform
    const int lane = tid & 31;

    if (wave < 2) {
        // ----- WMMA compute phase: wave w owns row-tile w (rows w*16..w*16+15)
        const int sub  = lane >> 4;     // 0: K base 0, 1: K base +2 (ISA A layout)
        const int mloc = lane & 15;     // M for A, N for B/C

        // A operand: row lr = wave*16 + mloc  ->  (b, voice r) ; l = fi*8 + r
        const int lr = wave * 16 + mloc;
        const int b  = lr >> 3;
        const int r  = lr & 7;
        const float* qrow = q + ((size_t)(b * NH + h) * LDIM + (fi * NV + r)) * EDIM;

        v2f a[4];
#pragma unroll
        for (int k = 0; k < 4; ++k) {
            const int e0 = 4 * k + 2 * sub;   // lanes 0-15: K=e0,e0+1 ; lanes 16-31: +2
            a[k].x = qrow[e0];
            a[k].y = qrow[e0 + 1];
        }

        // 7 column tiles: c=0..3 -> r_voice cols (vi*8+vo, 64), c=4..6 -> time cols (fj, 48)
#pragma unroll
        for (int c = 0; c < 7; ++c) {
            const int colg = c * 16 + mloc;   // column index within B
            v8f acc = {};
#pragma unroll
            for (int k = 0; k < 4; ++k) {
                const int e0 = 4 * k + 2 * sub;
                v2f bop;
                if (c < 4) {
                    // B[e][vi*8+vo] = r_voice[vi,vo,e,h] ; flat ((vi*8+vo)*16+e)*8+h
                    const size_t base = ((size_t)colg * EDIM + e0) * NH + h;
                    bop.x = r_voice[base];
                    bop.y = r_voice[base + NH];
                } else {
                    const int fj = colg - 64;
                    const int d  = fi - fj;
                    const float* src = (d >= 0) ? e_past : e_future;
                    const int ad = (d >= 0) ? d : -d;
                    const size_t base = ((size_t)ad * EDIM + e0) * NH + h;
                    bop.x = src[base];
                    bop.y = src[base + NH];
                }
                // D = A(16x4) * B(4x16) + C, fp32 — exact w.r.t. fp32 reference math
                acc = __builtin_amdgcn_wmma_f32_16x16x4_f32(
                    false, a[k], false, bop, (short)0, acc, false, false);
            }
            // C/D layout: VGPR v -> row v + 8*sub (within tile), col mloc
#pragma unroll
            for (int v = 0; v < 8; ++v) {
                lds[(wave * 16 + sub * 8 + v) * STR + c * 16 + mloc] = acc[v];
            }
        }
    }

    __syncthreads();

    // ----- Broadcast-store phase: 32 rows x 384 cols = 12288 floats = 3072 float4
    for (int idx = tid; idx < (32 * LDIM) / 4; idx += 256) {
        const int row = idx / (LDIM / 4);       // local row 0..31
        const int c4  = idx % (LDIM / 4);
        const int m0  = c4 * 4;
        const int b   = row >> 3;
        const int r   = row & 7;                // voice of l  (= l % 8)
        const int g   = b * NH + h;
        const int l   = fi * NV + r;
        const float* Prow = &lds[row * STR];

        v4f val;
#pragma unroll
        for (int j = 0; j < 4; ++j) {
            const int m = m0 + j;
            val[j] = Prow[r * 8 + (m & 7)]      // voice term: P1[row, (l%8)*8 + m%8]
                   + Prow[64 + (m >> 3)];       // time term:  P2[row, m/8]
        }
        *(v4f*)(out + ((size_t)g * LDIM + l) * LDIM + m0) = val;
    }
}

extern "C" void kernel_launch(void* const* d_in, const int* in_sizes, int n_in,
                              void* d_out, int out_size, void* d_ws, size_t ws_size,
                              hipStream_t stream) {
    (void)in_sizes; (void)n_in; (void)out_size; (void)d_ws; (void)ws_size;
    const float* q        = (const float*)d_in[0];
    // d_in[1] = flipped_masks (unused by forward)
    const float* r_voice  = (const float*)d_in[2];
    const float* e_past   = (const float*)d_in[3];
    const float* e_future = (const float*)d_in[4];
    float* out            = (float*)d_out;

    relattn_fused<<<dim3(NH * NF), dim3(256), 0, stream>>>(
        q, r_voice, e_past, e_future, out);
}